// AvgPoolingLayer_56599079027374
// MI455X (gfx1250) — compile-verified
//
#include <hip/hip_runtime.h>

// -----------------------------------------------------------------------------
// 2x2 avg-pool + per-channel affine on (16, 64, 512, 512) f32.
//
// Bandwidth-bound (0.25 FLOP/byte): 1 GiB in + 256 MiB out -> ~57us floor at
// 23.3 TB/s. CDNA5 path: GLOBAL_LOAD_ASYNC_TO_LDS_B128 (ASYNCcnt) staging with
// shared immediate offsets (INST_OFFSET applies to BOTH global and LDS
// addresses per ISA pseudocode), s_wait_asynccnt drain, NT hints both ways.
// -----------------------------------------------------------------------------

typedef float v4f __attribute__((ext_vector_type(4)));

#define C_CH   64
#define S_IN   512
#define H_OUT  256
#define W_OUT  256
#define W4     (W_OUT / 4)      // 64 float4 per output row
#define BLOCK  256              // 8 wave32 per block

// Per-wave LDS footprint: row0 [base, base+1KB), row1 [base+2048, base+3072)
// -> 3KB stride per wave, 8 waves = 24KB per block (~10 blocks/WGP resident).
#define WAVE_LDS_FLOATS 768     // 3KB / 4

__global__ __launch_bounds__(BLOCK)
void avgpool_affine_async(const float* __restrict__ x,
                          const float* __restrict__ weight,
                          const float* __restrict__ bias,
                          float* __restrict__ out,
                          long long n4)
{
    __shared__ float smem[8 * WAVE_LDS_FLOATS];   // 24 KB

    const int tid  = threadIdx.x;
    const int lane = tid & 31;
    const int wv   = tid >> 5;
    const long long i4 = (long long)blockIdx.x * BLOCK + tid;
    if (i4 >= n4) return;

    // Decompose output float4 index: (b*C + c, h, w4)
    const int       w4 = (int)(i4 & (W4 - 1));         // % 64
    const long long t  = i4 >> 6;                      // / 64
    const int       h  = (int)(t & (H_OUT - 1));       // % 256
    const long long bc = t >> 8;                       // b*C + c
    const int       c  = (int)(bc & (C_CH - 1));

    // Single global base per thread: row 2h, cols [8*w4, 8*w4+8).
    // The other three 16B chunks are reached via immediate offsets:
    //   +16 (row0 cols 4..7), +2048 (row 2h+1), +2064.
    const float* g0 = x + ((bc * S_IN) + 2 * h) * (long long)S_IN + (w4 << 3);

    // Single LDS base per thread (byte address = low 32 bits of flat shared
    // pointer; LDS aperture keeps the byte offset in addr[31:0]). The same
    // immediate offsets place row1 data at base+2048 inside this wave's 3KB
    // region — no cross-thread collision since lane*32 + 2064 + 16 <= 3072.
    const int      base = wv * WAVE_LDS_FLOATS + lane * 8;   // floats (row0)
    const unsigned lds  = (unsigned)(unsigned long long)&smem[base];

    // CDNA5 async global -> LDS copies (ASYNCcnt), read-once input -> NT.
    asm volatile(
        "global_load_async_to_lds_b128 %0, %1, off th:TH_LOAD_NT\n\t"
        "global_load_async_to_lds_b128 %0, %1, off offset:16 th:TH_LOAD_NT\n\t"
        "global_load_async_to_lds_b128 %0, %1, off offset:2048 th:TH_LOAD_NT\n\t"
        "global_load_async_to_lds_b128 %0, %1, off offset:2064 th:TH_LOAD_NT"
        :: "v"(lds), "v"(g0) : "memory");

    // Tiny per-channel params (L2-resident) overlap with the async copies.
    const float wc = weight[c];
    const float bb = bias[c];

    // Drain this wave's async copies; each lane reads only its own bytes,
    // so the per-wave ASYNCcnt wait is the full required fence (no barrier).
    asm volatile("s_wait_asynccnt 0" ::: "memory");

    const v4f a0 = *(const v4f*)&smem[base];            // row 2h,   cols 0..3
    const v4f a1 = *(const v4f*)&smem[base + 4];        // row 2h,   cols 4..7
    const v4f b0 = *(const v4f*)&smem[base + 512];      // row 2h+1, cols 0..3
    const v4f b1 = *(const v4f*)&smem[base + 516];      // row 2h+1, cols 4..7

    v4f s;
    s.x = (a0.x + a0.y) + (b0.x + b0.y);
    s.y = (a0.z + a0.w) + (b0.z + b0.w);
    s.z = (a1.x + a1.y) + (b1.x + b1.y);
    s.w = (a1.z + a1.w) + (b1.z + b1.w);

    const float sc = 0.25f * wc;        // fold pool divisor into the scale
    v4f r = s * sc + bb;

    // Write-once output: nontemporal 128-bit store.
    __builtin_nontemporal_store(r, (v4f*)(out + (i4 << 2)));
}

extern "C" void kernel_launch(void* const* d_in, const int* in_sizes, int n_in,
                              void* d_out, int out_size, void* d_ws, size_t ws_size,
                              hipStream_t stream)
{
    const float* x      = (const float*)d_in[0];
    const float* weight = (const float*)d_in[1];
    const float* bias   = (const float*)d_in[2];
    float*       out    = (float*)d_out;

    const long long n4 = (long long)out_size / 4;          // 16,777,216 float4s
    const long long blocks = (n4 + BLOCK - 1) / BLOCK;     // 65,536 blocks

    avgpool_affine_async<<<dim3((unsigned)blocks), dim3(BLOCK), 0, stream>>>(
        x, weight, bias, out, n4);
}